// InteractiveAttn_11407433138685
// MI455X (gfx1250) — compile-verified
//
#include <hip/hip_runtime.h>
#include <math.h>

#define SEQ 8192
#define HID 2048
#define KSPLIT 4
#define KSLICE (HID / KSPLIT)  // 512

typedef __attribute__((ext_vector_type(2))) float v2f;
typedef __attribute__((ext_vector_type(8))) float v8f;

// ---------------------------------------------------------------------------
// Kernel 1: partial column sums of  v[j] = sum_i hidden[i] * W_qh[i][j]
// grid (HID/256, 16), block 256.  vpart[chunk][j] over 128-row chunks.
// ---------------------------------------------------------------------------
__global__ __launch_bounds__(256) void vpart_kernel(
    const float* __restrict__ W_qh, const float* __restrict__ hidden,
    float* __restrict__ vpart) {
  const int j = blockIdx.x * 256 + threadIdx.x;
  const int chunk = blockIdx.y;
  const int i0 = chunk * (HID / 16);
  float acc = 0.f;
#pragma unroll 8
  for (int i = 0; i < HID / 16; ++i) {
    acc = fmaf(hidden[i0 + i], W_qh[(size_t)(i0 + i) * HID + j], acc);
  }
  vpart[chunk * HID + j] = acc;
}

// ---------------------------------------------------------------------------
// Kernel 2: w[j] = (W_pq[j,:].pool + b_pq[j]) * v[j]
// one wave per output row j; lane-strided dot + shuffle reduction.
// ---------------------------------------------------------------------------
__global__ __launch_bounds__(256) void wvec_kernel(
    const float* __restrict__ W_pq, const float* __restrict__ pool,
    const float* __restrict__ b_pq, const float* __restrict__ vpart,
    float* __restrict__ wvec) {
  const int lane = threadIdx.x & 31;
  const int wave = threadIdx.x >> 5;
  const int j = blockIdx.x * 8 + wave;
  const float* row = W_pq + (size_t)j * HID;
  float p = 0.f;
#pragma unroll 8
  for (int k = lane; k < HID; k += 32) p = fmaf(row[k], pool[k], p);
#pragma unroll
  for (int off = 16; off > 0; off >>= 1) p += __shfl_down(p, off, 32);
  if (lane == 0) {
    float m = p + b_pq[j];
    float v = 0.f;
#pragma unroll
    for (int c = 0; c < 16; ++c) v += vpart[c * HID + j];
    wvec[j] = m * v;
  }
}

// ---------------------------------------------------------------------------
// Kernel 3 (WMMA): partial energies via V_WMMA_F32_16X16X4_F32.
// grid (SEQ/64, KSPLIT), block 128 (4 waves x 16 rows).  Each block covers
// 64 seq rows over a 512-wide K slice -> 512 blocks / 2048 waves for
// bandwidth saturation.  A-tile: lane L<16 -> enc[row+L, k..k+1] (b64),
// lane L+16 -> enc[row+L, k+2..k+3]  (32-bit 16x4 A layout).  B-tile:
// w[k..k+3] broadcast across 16 columns from LDS; column 0 of D is the dot.
// C layout: lane 0 holds rows 0..7 in acc[0..7], lane 16 rows 8..15.
// ---------------------------------------------------------------------------
__global__ __launch_bounds__(128) void energies_kernel(
    const float* __restrict__ enc, const float* __restrict__ wvec,
    float* __restrict__ epart) {
  __shared__ float lw[KSLICE];
  const int k0 = blockIdx.y * KSLICE;
  for (int i = threadIdx.x; i < KSLICE; i += 128) lw[i] = wvec[k0 + i];
  __syncthreads();

  const int lane = threadIdx.x & 31;
  const int wave = threadIdx.x >> 5;
  const int row0 = blockIdx.x * 64 + wave * 16;
  const int mrow = lane & 15;
  const int koff = (lane >> 4) * 2;  // lanes 16-31 carry K+2, K+3

  const float* rp = enc + (size_t)(row0 + mrow) * HID + k0 + koff;
  v8f acc = {};
#pragma unroll 8
  for (int k = 0; k < KSLICE; k += 4) {
    v2f a = *(const v2f*)(rp + k);
    v2f b = *(const v2f*)(&lw[k + koff]);
    acc = __builtin_amdgcn_wmma_f32_16x16x4_f32(
        /*neg_a=*/false, a, /*neg_b=*/false, b,
        /*c_mod=*/(short)0, acc, /*reuse_a=*/false, /*reuse_b=*/false);
  }
  // Column 0 of D: lane 0 -> rows row0+0..7 in acc[0..7]; lane 16 -> +8..15.
  if ((lane & 15) == 0) {
    const int base = blockIdx.y * SEQ + row0 + (lane >> 4) * 8;
#pragma unroll
    for (int i = 0; i < 8; ++i) epart[base + i] = acc[i];
  }
}

// ---------------------------------------------------------------------------
// Kernel 4: sum KSPLIT partials + softmax over 8192.  One block, 1024 thr.
// ---------------------------------------------------------------------------
__global__ __launch_bounds__(1024) void softmax_kernel(
    const float* __restrict__ e, float* __restrict__ out) {
  __shared__ float red[32];
  const int t = threadIdx.x;
  const int lane = t & 31;
  const int wave = t >> 5;

  float vals[8];
  float mx = -INFINITY;
#pragma unroll
  for (int i = 0; i < 8; ++i) {
    const int idx = t + i * 1024;
    float v = 0.f;
#pragma unroll
    for (int p = 0; p < KSPLIT; ++p) v += e[p * SEQ + idx];
    vals[i] = v;
    mx = fmaxf(mx, v);
  }
#pragma unroll
  for (int off = 16; off > 0; off >>= 1) mx = fmaxf(mx, __shfl_xor(mx, off, 32));
  if (lane == 0) red[wave] = mx;
  __syncthreads();
  if (t < 32) {
    float r = red[t];
#pragma unroll
    for (int off = 16; off > 0; off >>= 1) r = fmaxf(r, __shfl_xor(r, off, 32));
    if (t == 0) red[0] = r;
  }
  __syncthreads();
  mx = red[0];
  __syncthreads();  // done reading red[0] before reuse

  float s = 0.f;
#pragma unroll
  for (int i = 0; i < 8; ++i) {
    vals[i] = __expf(vals[i] - mx);
    s += vals[i];
  }
#pragma unroll
  for (int off = 16; off > 0; off >>= 1) s += __shfl_xor(s, off, 32);
  if (lane == 0) red[wave] = s;
  __syncthreads();
  if (t < 32) {
    float r = red[t];
#pragma unroll
    for (int off = 16; off > 0; off >>= 1) r += __shfl_xor(r, off, 32);
    if (t == 0) red[0] = r;
  }
  __syncthreads();
  const float inv = 1.0f / red[0];
#pragma unroll
  for (int i = 0; i < 8; ++i) out[t + i * 1024] = vals[i] * inv;
}

// ---------------------------------------------------------------------------
// inputs: 0 hidden(2048) 1 encoder(8192*2048) 2 pool(2048)
//         3 W_pq(2048*2048) 4 b_pq(2048) 5 W_qh(2048*2048) 6 b_qh(2048)
// b_qh contributes a constant to every energy -> softmax-invariant, dropped.
// ws layout (floats): [0,2048) w | [2048,34816) vpart, later reused as
// epart (KSPLIT*8192 = 32768 floats) once vpart is consumed (stream order).
// ---------------------------------------------------------------------------
extern "C" void kernel_launch(void* const* d_in, const int* in_sizes, int n_in,
                              void* d_out, int out_size, void* d_ws, size_t ws_size,
                              hipStream_t stream) {
  const float* hidden = (const float*)d_in[0];
  const float* enc    = (const float*)d_in[1];
  const float* pool   = (const float*)d_in[2];
  const float* W_pq   = (const float*)d_in[3];
  const float* b_pq   = (const float*)d_in[4];
  const float* W_qh   = (const float*)d_in[5];
  float* ws    = (float*)d_ws;
  float* wvec  = ws;         // 2048 floats
  float* vpart = ws + HID;   // 16*2048 floats
  float* epart = ws + HID;   // KSPLIT*8192 floats (aliases vpart, safe)
  float* out   = (float*)d_out;

  vpart_kernel<<<dim3(HID / 256, 16), 256, 0, stream>>>(W_qh, hidden, vpart);
  wvec_kernel<<<HID / 8, 256, 0, stream>>>(W_pq, pool, b_pq, vpart, wvec);
  energies_kernel<<<dim3(SEQ / 64, KSPLIT), 128, 0, stream>>>(enc, wvec, epart);
  softmax_kernel<<<1, 1024, 0, stream>>>(epart, out);
}